// PointerNetworkAction_11175504904503
// MI455X (gfx1250) — compile-verified
//
#include <hip/hip_runtime.h>
#include <hip/hip_bf16.h>
#include <math.h>

#define B_  32
#define L_  2048
#define D_  256
#define H_  64
#define Y_  256

typedef float v2f __attribute__((ext_vector_type(2)));
typedef float v8f __attribute__((ext_vector_type(8)));

// ---------------------------------------------------------------------------
// y_t[b,h] = sum_d yvec[b,d] * w2[act][d,h] + b2[act][h]    (B x 64)
// ---------------------------------------------------------------------------
__global__ __launch_bounds__(64)
void yt_kernel(const float* __restrict__ yvec,   // (B, 256)
               const float* __restrict__ w2,     // (A, 256, 64)
               const float* __restrict__ b2,     // (A, 64)
               const int*   __restrict__ actions,
               float* __restrict__ yt)           // (B, 64)
{
    const int b = blockIdx.x;
    const int h = threadIdx.x;
    const int act = actions[b];
    const float* w  = w2 + (size_t)act * D_ * H_;
    const float* yv = yvec + (size_t)b * D_;
    float acc = b2[act * H_ + h];
    #pragma unroll 8
    for (int d = 0; d < D_; ++d)
        acc = fmaf(yv[d], w[d * H_ + h], acc);
    yt[b * H_ + h] = acc;
}

// ---------------------------------------------------------------------------
// x_t[b,l,h] = sum_d x[b,l,d] * w1[act][d,h] + b1[act][h]
// One block = 128 threads = 4 waves; block covers 16 L-rows x all 64 H cols.
// Each wave owns a 16x16 tile and runs 64 V_WMMA_F32_16X16X4_F32 steps.
// ---------------------------------------------------------------------------
__global__ __launch_bounds__(128)
void xt_gemm_kernel(const float* __restrict__ x,    // (B, L, D)
                    const float* __restrict__ w1,   // (A, D, H)
                    const float* __restrict__ b1,   // (A, H)
                    const int*   __restrict__ actions,
                    float* __restrict__ xt)         // (B, L, H)
{
    const int b    = blockIdx.y;
    const int l0   = blockIdx.x * 16;
    const int wave = threadIdx.x >> 5;    // 0..3 -> H tile
    const int lane = threadIdx.x & 31;
    const int h0   = wave * 16;
    const int half = lane >> 4;           // 0 or 1
    const int mn   = lane & 15;           // M for A, N for B

    const int act = actions[b];
    const float* xrow = x + ((size_t)b * L_ + (l0 + mn)) * D_ + 2 * half;
    const float* wcol = w1 + (size_t)act * D_ * H_ + (size_t)(2 * half) * H_ + (h0 + mn);

    // bias depends only on N -> preload into all C rows
    const float bias = b1[act * H_ + h0 + mn];
    v8f c;
    #pragma unroll
    for (int i = 0; i < 8; ++i) c[i] = bias;

    #pragma unroll 4
    for (int k = 0; k < D_; k += 4) {
        // A fragment: a[v] = x[m][k + 2*half + v]  (consecutive -> float2)
        v2f a = *(const v2f*)(xrow + k);
        // B fragment: bf[v] = w1a[k + 2*half + v][n]
        v2f bf;
        bf[0] = wcol[(size_t)k * H_];
        bf[1] = wcol[(size_t)(k + 1) * H_];
        c = __builtin_amdgcn_wmma_f32_16x16x4_f32(
                /*neg_a=*/false, a, /*neg_b=*/false, bf,
                /*c_mod=*/(short)0, c, /*reuse_a=*/false, /*reuse_b=*/false);
    }

    // C/D layout: VGPR i -> row (i + 8*half), col mn
    float* drow = xt + ((size_t)b * L_ + (l0 + 8 * half)) * H_ + h0 + mn;
    #pragma unroll
    for (int i = 0; i < 8; ++i)
        drow[(size_t)i * H_] = c[i];
}

// ---------------------------------------------------------------------------
// scores[b,l] = sum_h tanh(x_t[b,l,h] + y_t[b,h]) * v[act][h]; mask; softmax
// over L within one block (1024 threads, 2 rows/thread). Writes probs.
// ---------------------------------------------------------------------------
__global__ __launch_bounds__(1024)
void score_softmax_kernel(const float* __restrict__ xt,      // (B, L, H)
                          const float* __restrict__ yt,      // (B, H)
                          const float* __restrict__ v,       // (A, H)
                          const int*   __restrict__ actions,
                          const unsigned char* __restrict__ x_mask, // (B, L)
                          float* __restrict__ probs)         // (B, L)
{
    const int b   = blockIdx.x;
    const int tid = threadIdx.x;
    __shared__ float sh_yv[2 * H_];   // [0:64) = y_t, [64:128) = v[act]
    __shared__ float red[1024];

    const int act = actions[b];
    if (tid < H_)            sh_yv[tid] = yt[b * H_ + tid];
    else if (tid < 2 * H_)   sh_yv[tid] = v[act * H_ + (tid - H_)];
    __syncthreads();

    float s[2];
    #pragma unroll
    for (int j = 0; j < 2; ++j) {
        const int l = tid + j * 1024;
        const float* row = xt + ((size_t)b * L_ + l) * H_;
        float acc = 0.0f;
        #pragma unroll 8
        for (int h = 0; h < H_; ++h)
            acc = fmaf(tanhf(row[h] + sh_yv[h]), sh_yv[H_ + h], acc);
        if (x_mask[(size_t)b * L_ + l]) acc = -INFINITY;
        s[j] = acc;
    }

    // block max
    red[tid] = fmaxf(s[0], s[1]);
    __syncthreads();
    for (int off = 512; off > 0; off >>= 1) {
        if (tid < off) red[tid] = fmaxf(red[tid], red[tid + off]);
        __syncthreads();
    }
    const float mx = red[0];
    __syncthreads();

    const float e0 = expf(s[0] - mx);
    const float e1 = expf(s[1] - mx);
    red[tid] = e0 + e1;
    __syncthreads();
    for (int off = 512; off > 0; off >>= 1) {
        if (tid < off) red[tid] += red[tid + off];
        __syncthreads();
    }
    const float inv = 1.0f / red[0];

    probs[(size_t)b * L_ + tid]        = e0 * inv;
    probs[(size_t)b * L_ + tid + 1024] = e1 * inv;
}

// ---------------------------------------------------------------------------
// pool[b,d] = sum_l x[b,l,d] * probs[b,l]
// ---------------------------------------------------------------------------
__global__ __launch_bounds__(256)
void attn_pool_kernel(const float* __restrict__ x,      // (B, L, D)
                      const float* __restrict__ probs,  // (B, L)
                      float* __restrict__ pool)         // (B, D)
{
    const int b = blockIdx.x;
    const int d = threadIdx.x;
    const float* xb = x + (size_t)b * L_ * D_ + d;
    const float* p  = probs + (size_t)b * L_;
    float acc = 0.0f;
    #pragma unroll 8
    for (int l = 0; l < L_; ++l)
        acc = fmaf(xb[(size_t)l * D_], p[l], acc);
    pool[b * D_ + d] = acc;
}

// ---------------------------------------------------------------------------
// SRU cell: u = pool @ w_sru (only first 2Y cols needed)
// state = sigmoid(u[:,Y:2Y] + b_sru[:Y]) * c0 + (1 - f) * u[:, :Y]
// ---------------------------------------------------------------------------
__global__ __launch_bounds__(256)
void sru_kernel(const float* __restrict__ pool,   // (B, D)
                const float* __restrict__ c0,     // (B, Y)
                const float* __restrict__ w_sru,  // (D, 3Y)
                const float* __restrict__ b_sru,  // (2Y)
                float* __restrict__ state)        // (B, Y)
{
    const int b = blockIdx.x;
    const int y = threadIdx.x;
    __shared__ float pb[D_];
    pb[y] = pool[b * D_ + y];
    __syncthreads();

    float u1 = 0.0f, u2 = 0.0f;
    #pragma unroll 4
    for (int d = 0; d < D_; ++d) {
        const float pv = pb[d];
        u1 = fmaf(pv, w_sru[(size_t)d * (3 * Y_) + y], u1);
        u2 = fmaf(pv, w_sru[(size_t)d * (3 * Y_) + Y_ + y], u2);
    }
    const float f = 1.0f / (1.0f + expf(-(u2 + b_sru[y])));
    state[b * Y_ + y] = f * c0[b * Y_ + y] + (1.0f - f) * u1;
}

// ---------------------------------------------------------------------------
extern "C" void kernel_launch(void* const* d_in, const int* in_sizes, int n_in,
                              void* d_out, int out_size, void* d_ws, size_t ws_size,
                              hipStream_t stream) {
    const float*         x       = (const float*)d_in[0];
    const unsigned char* x_mask  = (const unsigned char*)d_in[1];  // all-false bools
    const float*         c0      = (const float*)d_in[2];
    const int*           actions = (const int*)d_in[3];
    const float*         w1      = (const float*)d_in[4];
    const float*         b1      = (const float*)d_in[5];
    const float*         w2      = (const float*)d_in[6];
    const float*         b2      = (const float*)d_in[7];
    const float*         v       = (const float*)d_in[8];
    const float*         w_sru   = (const float*)d_in[9];
    const float*         b_sru   = (const float*)d_in[10];

    float* out     = (float*)d_out;
    float* s_probs = out;                 // (B, L)
    float* e_probs = out + (size_t)B_ * L_;

    float* ws    = (float*)d_ws;
    float* xt    = ws;                               // B*L*H = 4,194,304 floats
    float* yt_s  = xt   + (size_t)B_ * L_ * H_;      // B*H
    float* yt_e  = yt_s + (size_t)B_ * H_;           // B*H
    float* pool  = yt_e + (size_t)B_ * H_;           // B*D
    float* state = pool + (size_t)B_ * D_;           // B*Y

    // s-pass gate vector from c0
    yt_kernel<<<B_, H_, 0, stream>>>(c0, w2, b2, actions, yt_s);

    // shared big GEMM: x_t (used by BOTH attention passes)
    xt_gemm_kernel<<<dim3(L_ / 16, B_), 128, 0, stream>>>(x, w1, b1, actions, xt);

    // s logits -> softmax -> output 0
    score_softmax_kernel<<<B_, 1024, 0, stream>>>(xt, yt_s, v, actions, x_mask, s_probs);

    // attention-pooled context
    attn_pool_kernel<<<B_, D_, 0, stream>>>(x, s_probs, pool);

    // SRU cell -> new state
    sru_kernel<<<B_, Y_, 0, stream>>>(pool, c0, w_sru, b_sru, state);

    // e-pass gate vector from state
    yt_kernel<<<B_, H_, 0, stream>>>(state, w2, b2, actions, yt_e);

    // e logits -> softmax -> output 1
    score_softmax_kernel<<<B_, 1024, 0, stream>>>(xt, yt_e, v, actions, x_mask, e_probs);
}